// Update_method_3504693313970
// MI455X (gfx1250) — compile-verified
//
#include <hip/hip_runtime.h>

#define B_SZ   64
#define DIN    1024
#define DD     768
#define CC     1000
#define CPAD   1008      // padded LDS row (16B-aligned length >= CC)
#define EPSILON 0.1f
#define NUM_SIM 3

typedef __attribute__((ext_vector_type(2))) float v2f;
typedef __attribute__((ext_vector_type(8))) float v8f;

// ---------------------------------------------------------------------------
// fp32 WMMA GEMM:  C[M,N] = A[M,K] * B + (bias over N)
//   A row-major, leading dim lda.  B element (k,n) at Bm[k*ldbK + n*ldbN].
//   BK1 == true  -> ldbK == 1  (A*B^T style, K contiguous -> b64 loads)
//   BK1 == false -> ldbN == 1  (A*B style, K strided)
// One wave computes a 16x64 tile: 4 accumulators share each A fragment.
// Fragment layout (ISA 7.12.2, wave32):
//   A 16x4 : lane l&15 = row, half h = l>>4 -> regs hold K = 2h, 2h+1
//   B 4x16 : lane l&15 = col, half h        -> regs hold K = 2h, 2h+1
//   C/D    : reg r, half h -> row = r + 8h, col = lane&15
// The k-loop is software-pipelined: fragments for step k+4 are loaded while
// the four WMMAs for step k execute, so the 5 in-flight loads overlap matrix
// work (partial s_wait_loadcnt instead of full drains). Out-of-range lanes
// clamp indices (reads stay in-bounds; garbage lands only in D rows/cols the
// store guards discard), so the loop is branch-free and EXEC stays all-ones
// around every WMMA. K must be a multiple of 4 (1024 / 768 / 1000 all are).
// ---------------------------------------------------------------------------
template <bool BK1>
__device__ __forceinline__ v2f load_bfrag(const float* __restrict__ p, int ka, int ldbK)
{
    v2f b;
    if (BK1) {
        b = *(const v2f*)(p + ka);                    // global_load_b64
    } else {
        b.x = p[(size_t)ka * ldbK];
        b.y = p[(size_t)(ka + 1) * ldbK];
    }
    return b;
}

template <bool BK1>
__global__ __launch_bounds__(32)
void gemm_wmma_f32(const float* __restrict__ A, const float* __restrict__ Bm,
                   const float* __restrict__ bias, float* __restrict__ Cm,
                   int M, int N, int K, int lda, int ldbK, int ldbN, int ldc)
{
    const int lane = threadIdx.x;
    const int half = lane >> 4;
    const int l    = lane & 15;
    const int m0   = blockIdx.y * 16;
    const int n0   = blockIdx.x * 64;

    const int am = min(m0 + l, M - 1);
    int bn[4];
#pragma unroll
    for (int t = 0; t < 4; ++t) bn[t] = min(n0 + 16 * t + l, N - 1);

    const float* __restrict__ Arow = A + (size_t)am * lda;
    const float* __restrict__ Bcol[4];
#pragma unroll
    for (int t = 0; t < 4; ++t)
        Bcol[t] = BK1 ? (Bm + (size_t)bn[t] * ldbN) : (Bm + bn[t]);

    v8f acc[4] = {};

    // ---- software pipeline: preload k = 0 fragments -----------------------
    int ka   = 2 * half;
    v2f a_c  = *(const v2f*)(Arow + ka);
    v2f b_c[4];
#pragma unroll
    for (int t = 0; t < 4; ++t) b_c[t] = load_bfrag<BK1>(Bcol[t], ka, ldbK);

#pragma unroll 2
    for (int k = 0; k < K; k += 4) {
        // issue next step's loads first (last iter re-loads k: uniform & safe)
        const int kn  = (k + 4 < K) ? (k + 4) : k;
        const int kan = kn + 2 * half;
        v2f a_n = *(const v2f*)(Arow + kan);
        v2f b_n[4];
#pragma unroll
        for (int t = 0; t < 4; ++t) b_n[t] = load_bfrag<BK1>(Bcol[t], kan, ldbK);

        // compute on current fragments while next loads are in flight
#pragma unroll
        for (int t = 0; t < 4; ++t)
            acc[t] = __builtin_amdgcn_wmma_f32_16x16x4_f32(
                false, a_c, false, b_c[t], (short)0, acc[t], false, false);

        a_c = a_n;
#pragma unroll
        for (int t = 0; t < 4; ++t) b_c[t] = b_n[t];
    }

#pragma unroll
    for (int t = 0; t < 4; ++t) {
        const int cn = n0 + 16 * t + l;
        if (cn < N) {
            const float bv = bias ? bias[cn] : 0.f;
#pragma unroll
            for (int r = 0; r < 8; ++r) {
                const int cm = m0 + r + 8 * half;
                if (cm < M) Cm[(size_t)cm * ldc + cn] = acc[t][r] + bv;
            }
        }
    }
}

// ---------------------------------------------------------------------------
// Row softmax: one 128-thread block (4 wave32) per row.
// ---------------------------------------------------------------------------
__global__ __launch_bounds__(128)
void softmax_rows(const float* __restrict__ in, float* __restrict__ out, int Ccols)
{
    const int row  = blockIdx.x;
    const int tid  = threadIdx.x;
    const int lane = tid & 31, wid = tid >> 5;
    __shared__ float smax[4];
    __shared__ float ssum[4];
    const float* rp = in  + (size_t)row * Ccols;
    float*       op = out + (size_t)row * Ccols;

    float m = -3.402823466e38f;
    for (int c = tid; c < Ccols; c += 128) m = fmaxf(m, rp[c]);
    for (int off = 16; off; off >>= 1) m = fmaxf(m, __shfl_xor(m, off, 32));
    if (lane == 0) smax[wid] = m;
    __syncthreads();
    m = fmaxf(fmaxf(smax[0], smax[1]), fmaxf(smax[2], smax[3]));

    float s = 0.f;
    for (int c = tid; c < Ccols; c += 128) s += expf(rp[c] - m);
    for (int off = 16; off; off >>= 1) s += __shfl_xor(s, off, 32);
    if (lane == 0) ssum[wid] = s;
    __syncthreads();
    s = ssum[0] + ssum[1] + ssum[2] + ssum[3];
    const float inv = 1.f / s;

    for (int c = tid; c < Ccols; c += 128) op[c] = expf(rp[c] - m) * inv;
}

// ---------------------------------------------------------------------------
// Row sum of squares: out[row] = sum_c in[row,c]^2
// ---------------------------------------------------------------------------
__global__ __launch_bounds__(128)
void row_sumsq(const float* __restrict__ in, float* __restrict__ out, int Ccols)
{
    const int row  = blockIdx.x;
    const int tid  = threadIdx.x;
    const int lane = tid & 31, wid = tid >> 5;
    __shared__ float sred[4];
    const float* rp = in + (size_t)row * Ccols;

    float s = 0.f;
    for (int c = tid; c < Ccols; c += 128) { float v = rp[c]; s += v * v; }
    for (int off = 16; off; off >>= 1) s += __shfl_xor(s, off, 32);
    if (lane == 0) sred[wid] = s;
    __syncthreads();
    if (tid == 0) out[row] = sred[0] + sred[1] + sred[2] + sred[3];
}

// ---------------------------------------------------------------------------
// Row argmax (first-max tie-break like jnp.argmax): one block per row.
// ---------------------------------------------------------------------------
__global__ __launch_bounds__(128)
void argmax_rows(const float* __restrict__ in, int* __restrict__ out, int Ccols)
{
    const int row  = blockIdx.x;
    const int tid  = threadIdx.x;
    const int lane = tid & 31, wid = tid >> 5;
    __shared__ float sv[4];
    __shared__ int   si[4];
    const float* rp = in + (size_t)row * Ccols;

    float best = -3.402823466e38f;
    int   bidx = 0;
    for (int c = tid; c < Ccols; c += 128) {
        float v = rp[c];
        if (v > best) { best = v; bidx = c; }
    }
    for (int off = 16; off; off >>= 1) {
        float ov = __shfl_xor(best, off, 32);
        int   oi = __shfl_xor(bidx, off, 32);
        if (ov > best || (ov == best && oi < bidx)) { best = ov; bidx = oi; }
    }
    if (lane == 0) { sv[wid] = best; si[wid] = bidx; }
    __syncthreads();
    if (tid == 0) {
        for (int w = 1; w < 4; ++w)
            if (sv[w] > best || (sv[w] == best && si[w] < bidx)) { best = sv[w]; bidx = si[w]; }
        out[row] = bidx;
    }
}

// ---------------------------------------------------------------------------
// Async copy of one 1000-float WW row into LDS: 8 x b128 per wave
// (32 lanes x 16B = 512B per op). Uses ASYNCcnt; rows complete in order.
// ldsDst must be the LDS byte offset (low 32 bits of the generic pointer).
// ---------------------------------------------------------------------------
__device__ __forceinline__
void async_copy_row(const float* __restrict__ src, unsigned ldsDst, int lane)
{
#pragma unroll
    for (int i = 0; i < 8; ++i) {
        const int elt = i * 128 + lane * 4;               // float index, 16B-aligned
        const unsigned           lds = ldsDst + (unsigned)(elt * 4);
        const unsigned long long ga  = (unsigned long long)(size_t)(src + elt);
        asm volatile("global_load_async_to_lds_b128 %0, %1, off"
                     :: "v"(lds), "v"(ga) : "memory");
    }
}

// ---------------------------------------------------------------------------
// Final stage. One block of 4 wave32 per b. For each pseudo-label j:
//   s = EPS * ||anchors[j]-emb[b]|| / ||G[b]-W_fc[j]||   (expanded norms)
//   pred_mix = argmax_c ( logits[b,c] + s * (GW[b,c] - WW[j,c]) )
// Each wave double-buffers WW rows in LDS via GLOBAL_LOAD_ASYNC_TO_LDS_B128:
// row j+4 streams in while row j is argmax-scanned; s_wait_asynccnt 8 gates
// the flip (8 ops per row, in-order completion). logits/GW rows are staged in
// LDS once. Histogram pred_mix (+ predictions[b]) in LDS, then emit
// all(counts < NUM_SIM) as 0/1 float.
// ---------------------------------------------------------------------------
__global__ __launch_bounds__(128)
void simulate_and_flag(const float* __restrict__ logits, const float* __restrict__ GW,
                       const float* __restrict__ AE,     const float* __restrict__ WW,
                       const float* __restrict__ En2,    const float* __restrict__ Gn2,
                       const float* __restrict__ Wn2,    const float* __restrict__ An2,
                       const int* __restrict__ pred,     float* __restrict__ outFlags)
{
    const int b    = blockIdx.x;
    const int tid  = threadIdx.x;
    const int lane = tid & 31, wid = tid >> 5;   // 4 waves

    __shared__ int   counts[CC];
    __shared__ int   smax[4];
    __shared__ float lgs[CPAD];
    __shared__ float gws[CPAD];
    __shared__ __align__(16) float wbuf[4][2][CPAD];

    const float* lg = logits + (size_t)b * CC;
    const float* gw = GW     + (size_t)b * CC;
    const float* ae = AE     + (size_t)b * CC;

    for (int c = tid; c < CC; c += 128) {
        counts[c] = 0;
        lgs[c]    = lg[c];
        gws[c]    = gw[c];
    }
    __syncthreads();
    if (tid == 0) atomicAdd(&counts[pred[b]], 1);

    const float en2  = En2[b];
    const float gn2b = Gn2[b];

    const unsigned buf0 = (unsigned)(size_t)(&wbuf[wid][0][0]);
    const unsigned buf1 = (unsigned)(size_t)(&wbuf[wid][1][0]);

    // prologue: start streaming this wave's first row
    async_copy_row(WW + (size_t)wid * CC, buf0, lane);

    int cur = 0;
    for (int j = wid; j < CC; j += 4) {
        // issue next row into the other buffer (clamped dummy on last iter
        // to keep ASYNCcnt bookkeeping uniform)
        const int jn = (j + 4 < CC) ? (j + 4) : (CC - 1);
        async_copy_row(WW + (size_t)jn * CC, cur ? buf0 : buf1, lane);
        // previous 8 ops (row j) are complete once <= 8 remain outstanding
        asm volatile("s_wait_asynccnt 0x8" ::: "memory");

        float zn2 = fmaxf(An2[j] - 2.f * ae[j] + en2, 0.f);
        float gn2 = fmaxf(gn2b - 2.f * gw[j] + Wn2[j], 1e-30f);
        float s   = EPSILON * sqrtf(zn2) * rsqrtf(gn2);

        const float* wb = &wbuf[wid][cur][0];
        float best = -3.402823466e38f;
        int   bidx = 0;
        for (int c = lane; c < CC; c += 32) {
            float val = lgs[c] + s * (gws[c] - wb[c]);   // ds_load traffic
            if (val > best) { best = val; bidx = c; }
        }
        for (int off = 16; off; off >>= 1) {
            float ov = __shfl_xor(best, off, 32);
            int   oi = __shfl_xor(bidx, off, 32);
            if (ov > best || (ov == best && oi < bidx)) { best = ov; bidx = oi; }
        }
        if (lane == 0) atomicAdd(&counts[bidx], 1);
        cur ^= 1;
    }
    __syncthreads();

    int mx = 0;
    for (int c = tid; c < CC; c += 128) mx = max(mx, counts[c]);
    for (int off = 16; off; off >>= 1) mx = max(mx, __shfl_xor(mx, off, 32));
    if (lane == 0) smax[wid] = mx;
    __syncthreads();
    if (tid == 0) {
        int m2 = max(max(smax[0], smax[1]), max(smax[2], smax[3]));
        outFlags[b] = (m2 < NUM_SIM) ? 1.0f : 0.0f;
    }
}

// ---------------------------------------------------------------------------
extern "C" void kernel_launch(void* const* d_in, const int* in_sizes, int n_in,
                              void* d_out, int out_size, void* d_ws, size_t ws_size,
                              hipStream_t stream)
{
    const float* x       = (const float*)d_in[0];   // [64,1024]
    const float* W_emb   = (const float*)d_in[1];   // [768,1024]
    const float* b_emb   = (const float*)d_in[2];   // [768]
    const float* W_fc    = (const float*)d_in[3];   // [1000,768]
    const float* b_fc    = (const float*)d_in[4];   // [1000]
    const float* anchors = (const float*)d_in[5];   // [1000,768]
    float* out = (float*)d_out;                     // [64] bool -> 0/1 float

    float* ws     = (float*)d_ws;
    float* emb    = ws;                      // 64*768
    float* logits = emb    + B_SZ * DD;      // 64*1000
    float* p      = logits + B_SZ * CC;      // 64*1000
    float* G      = p      + B_SZ * CC;      // 64*768
    float* GW     = G      + B_SZ * DD;      // 64*1000
    float* AE     = GW     + B_SZ * CC;      // 64*1000
    float* WW     = AE     + B_SZ * CC;      // 1000*1000
    float* En2    = WW     + CC * CC;        // 64
    float* Gn2    = En2    + B_SZ;           // 64
    float* Wn2    = Gn2    + B_SZ;           // 1000
    float* An2    = Wn2    + CC;             // 1000
    int*   pred   = (int*)(An2 + CC);        // 64

    const dim3 wblk(32);
    const int  n64C = (CC + 63) / 64;        // 16
    const int  n64D = (DD + 63) / 64;        // 12
    const int  mtC  = (CC + 15) / 16;        // 63

    // emb = x @ W_emb^T + b_emb        (B[k,n] = W_emb[n*DIN + k])
    gemm_wmma_f32<true><<<dim3(n64D, B_SZ / 16), wblk, 0, stream>>>(
        x, W_emb, b_emb, emb, B_SZ, DD, DIN, DIN, 1, DIN, DD);
    // logits = emb @ W_fc^T + b_fc
    gemm_wmma_f32<true><<<dim3(n64C, B_SZ / 16), wblk, 0, stream>>>(
        emb, W_fc, b_fc, logits, B_SZ, CC, DD, DD, 1, DD, CC);
    // p = softmax(logits)
    softmax_rows<<<B_SZ, 128, 0, stream>>>(logits, p, CC);
    // G = p @ W_fc                     (B[k,n] = W_fc[k*DD + n], K-strided)
    gemm_wmma_f32<false><<<dim3(n64D, B_SZ / 16), wblk, 0, stream>>>(
        p, W_fc, nullptr, G, B_SZ, DD, CC, CC, DD, 1, DD);
    // GW = G @ W_fc^T
    gemm_wmma_f32<true><<<dim3(n64C, B_SZ / 16), wblk, 0, stream>>>(
        G, W_fc, nullptr, GW, B_SZ, CC, DD, DD, 1, DD, CC);
    // AE = emb @ anchors^T
    gemm_wmma_f32<true><<<dim3(n64C, B_SZ / 16), wblk, 0, stream>>>(
        emb, anchors, nullptr, AE, B_SZ, CC, DD, DD, 1, DD, CC);
    // WW = W_fc @ W_fc^T
    gemm_wmma_f32<true><<<dim3(n64C, mtC), wblk, 0, stream>>>(
        W_fc, W_fc, nullptr, WW, CC, CC, DD, DD, 1, DD, CC);

    // squared row norms
    row_sumsq<<<B_SZ, 128, 0, stream>>>(emb,     En2, DD);
    row_sumsq<<<B_SZ, 128, 0, stream>>>(G,       Gn2, DD);
    row_sumsq<<<CC,   128, 0, stream>>>(W_fc,    Wn2, DD);
    row_sumsq<<<CC,   128, 0, stream>>>(anchors, An2, DD);

    // predictions = argmax(logits)
    argmax_rows<<<B_SZ, 128, 0, stream>>>(logits, pred, CC);

    // simulate pseudo-label mixes, histogram, emit all(counts < NUM_SIM)
    simulate_and_flag<<<B_SZ, 128, 0, stream>>>(
        logits, GW, AE, WW, En2, Gn2, Wn2, An2, pred, out);
}